// KaiserFilter_90091234001107
// MI455X (gfx1250) — compile-verified
//
#include <hip/hip_runtime.h>

// Fused separable 5-tap circular Kaiser filter (W then H) over [48][1024][1024] f32.
// Memory-bound: ~403 MB min traffic -> ~17us floor at 23.3 TB/s. Uses the CDNA5
// async global->LDS path (ASYNCcnt) for the tile loads.

#define IMG        1024
#define IMG_MASK   1023
#define TILE_H     32
#define TILE_W     64
#define IN_H       36              // TILE_H + 4 halo rows
#define IN_W       72              // TILE_W + 8 cols (halo 2 padded to 4 for b128 alignment)
#define MID_W      64
#define NTHREADS   256
#define NLOAD4     (IN_H * (IN_W / 4))          // 648 float4 async loads per block
#define SMEM_FLOATS (IN_H * IN_W + IN_H * MID_W) // 2592 + 2304 = 4896
#define SMEM_BYTES  (SMEM_FLOATS * 4)            // 19584 B

__global__ __launch_bounds__(NTHREADS)
void kaiser5x5_fused(const float* __restrict__ x,
                     const float* __restrict__ kern,
                     float* __restrict__ out)
{
    // Dynamic LDS only (no static LDS) => s_in starts at LDS byte offset 0,
    // which we rely on for the async-load destination addresses.
    extern __shared__ __align__(16) float smem[];
    float* s_in  = smem;                  // [IN_H][IN_W]
    float* s_mid = smem + IN_H * IN_W;    // [IN_H][MID_W]

    const int tid    = threadIdx.x;
    const int tile_c = blockIdx.x * TILE_W;   // multiple of 64
    const int tile_r = blockIdx.y * TILE_H;
    const int img    = blockIdx.z;

    const float* __restrict__ xin  = x   + (size_t)img * IMG * IMG;
    float*       __restrict__ xout = out + (size_t)img * IMG * IMG;

    // ---- Phase 0: async global -> LDS tile load with circular halo ----
    // Row i of s_in  <-> global row (tile_r - 2 + i) mod 1024
    // Col j of s_in  <-> global col (tile_c - 4 + j) mod 1024
    // tile_c-4 is 16B aligned and col starts are multiples of 4, so a float4
    // never straddles the circular wrap (1024 % 4 == 0).
    for (int k = tid; k < NLOAD4; k += NTHREADS) {
        int i  = k / (IN_W / 4);
        int j4 = k - i * (IN_W / 4);
        int j  = j4 * 4;
        int gr = (tile_r - 2 + i) & IMG_MASK;
        int gc = (tile_c - 4 + j) & IMG_MASK;
        unsigned long long gaddr =
            (unsigned long long)(xin + (size_t)gr * IMG + gc);
        unsigned laddr = (unsigned)((i * IN_W + j) * 4);   // LDS byte offset (base 0)
        asm volatile("global_load_async_to_lds_b128 %0, %1, off"
                     :
                     : "v"(laddr), "v"(gaddr)
                     : "memory");
    }

    // Load the 5 taps while async copies are in flight (L2-resident, tiny).
    float kk[5];
    #pragma unroll
    for (int t = 0; t < 5; ++t) kk[t] = kern[t];

    // Wait for this wave's async LDS writes, then sync the workgroup.
    asm volatile("s_wait_asynccnt 0" ::: "memory");
    __syncthreads();

    // ---- Phase 1: horizontal pass (rows keep their halo) ----
    // mid[i][j] = filtered value at global col tile_c+j, row tile_r-2+i.
    // s_in col (j+2+t) is global col tile_c + j + (t-2); FFT reference applies
    // k[2-d] at offset d, i.e. tap k[4-t] (symmetric window, but be exact).
    for (int k = tid; k < IN_H * TILE_W; k += NTHREADS) {   // 2304 = 9*256
        int i = k >> 6;
        int j = k & 63;
        const float* r = &s_in[i * IN_W + j + 2];
        float acc = kk[4] * r[0];
        acc += kk[3] * r[1];
        acc += kk[2] * r[2];
        acc += kk[1] * r[3];
        acc += kk[0] * r[4];
        s_mid[i * MID_W + j] = acc;
    }
    __syncthreads();

    // ---- Phase 2: vertical pass + coalesced float4 stores ----
    for (int k = tid; k < TILE_H * (TILE_W / 4); k += NTHREADS) {  // 512 = 2*256
        int i = k >> 4;
        int j = (k & 15) * 4;
        float4 acc = make_float4(0.f, 0.f, 0.f, 0.f);
        #pragma unroll
        for (int t = 0; t < 5; ++t) {
            float c = kk[4 - t];
            const float4 m = *(const float4*)&s_mid[(i + t) * MID_W + j];
            acc.x += c * m.x;
            acc.y += c * m.y;
            acc.z += c * m.z;
            acc.w += c * m.w;
        }
        *(float4*)&xout[(size_t)(tile_r + i) * IMG + tile_c + j] = acc;
    }
}

extern "C" void kernel_launch(void* const* d_in, const int* in_sizes, int n_in,
                              void* d_out, int out_size, void* d_ws, size_t ws_size,
                              hipStream_t stream) {
    (void)n_in; (void)out_size; (void)d_ws; (void)ws_size;
    const float* x    = (const float*)d_in[0];
    const float* kern = (const float*)d_in[1];
    float*       out  = (float*)d_out;

    const int nimg = in_sizes[0] / (IMG * IMG);   // 16*3 = 48
    dim3 grid(IMG / TILE_W, IMG / TILE_H, nimg);  // 16 x 32 x 48
    dim3 block(NTHREADS);
    kaiser5x5_fused<<<grid, block, SMEM_BYTES, stream>>>(x, kern, out);
}